// VortexBackbone_11209864643115
// MI455X (gfx1250) — compile-verified
//
#include <hip/hip_runtime.h>

// ---------------- problem dimensions (from reference) ----------------
#define B_    2
#define NCAM_ 6
#define J_    23
#define HM_   160
#define W_    640
#define H_    512
#define D_    50
#define D3_   (D_ * D_ * D_)

static constexpr long long HF_SIZE  = (long long)B_ * J_ * D3_;            // 5,750,000
static constexpr long long PAD_SIZE = (long long)B_ * NCAM_ * J_ * H_ * W_; // 90,439,680
static constexpr long long PTS_OFF  = HF_SIZE + PAD_SIZE;                   // 96,189,680

typedef __attribute__((ext_vector_type(2))) float v2f;
typedef __attribute__((ext_vector_type(8))) float v8f;

// ---------------------------------------------------------------------
// Kernel 1: fused zero + crop placement for heatmaps_padded.
// One float4 (b128) store per thread over the whole [B,NCAM,J,H,W] canvas.
// Branch is coherent: x0/y0 depend only on (b,cam).
// ---------------------------------------------------------------------
__global__ void __launch_bounds__(256)
place_kernel(const float* __restrict__ hm, const int* __restrict__ centerHM,
             float* __restrict__ out) {
    long long g = (long long)blockIdx.x * blockDim.x + threadIdx.x; // float4 index
    const long long total4 = PAD_SIZE / 4;
    if (g >= total4) return;

    int x4      = (int)(g % (W_ / 4));
    long long r = g / (W_ / 4);
    int y   = (int)(r % H_);
    int bcj = (int)(r / H_);
    int j   = bcj % J_;
    int bc  = bcj / J_;           // b*NCAM + cam

    int x0 = centerHM[bc * 2 + 0] / 2 - HM_ / 2;
    int y0 = centerHM[bc * 2 + 1] / 2 - HM_ / 2;

    float4 v = make_float4(0.f, 0.f, 0.f, 0.f);
    int x = x4 * 4;
    if ((unsigned)(y - y0) < (unsigned)HM_) {
        const float* src = hm + ((long long)(bc * J_ + j) * HM_ + (y - y0)) * HM_;
        int dx = x - x0;
        float vv[4] = {0.f, 0.f, 0.f, 0.f};
#pragma unroll
        for (int i = 0; i < 4; ++i) {
            int xx = dx + i;
            if ((unsigned)xx < (unsigned)HM_) vv[i] = src[xx];
        }
        v = make_float4(vv[0], vv[1], vv[2], vv[3]);
    }
    ((float4*)out)[g] = v;
}

// ---------------------------------------------------------------------
// Kernel 2: heatmap_final = softplus(heatmap_vol), vectorized float4.
// Stable form: max(x,0) + log1p(exp(-|x|)).
// ---------------------------------------------------------------------
__device__ __forceinline__ float softplus_f(float x) {
    return fmaxf(x, 0.f) + log1pf(expf(-fabsf(x)));
}

__global__ void __launch_bounds__(256)
softplus_kernel(const float4* __restrict__ in, float4* __restrict__ out, int n4) {
    int g = blockIdx.x * blockDim.x + threadIdx.x;
    if (g >= n4) return;
    float4 x = in[g];
    float4 y;
    y.x = softplus_f(x.x);
    y.y = softplus_f(x.y);
    y.z = softplus_f(x.z);
    y.w = softplus_f(x.w);
    out[g] = y;
}

// ---------------------------------------------------------------------
// Kernel 3: fused soft-argmax per (b,j). One pass accumulates
// {sum, sum*ix, sum*iy, sum*iz}; block reduction of the 256 partials is
// done with V_WMMA_F32_16X16X4_F32 (A = ones -> column sums), chaining C
// over 4 chunks of 64 partials. With A all-ones the total is invariant
// to the (k,n) placement inside B, so only the documented C/D row-0
// layout (lanes 0-15, vgpr 0) is relied upon.
// ---------------------------------------------------------------------
__global__ void __launch_bounds__(256)
reduce_kernel(const float* __restrict__ hf, const float* __restrict__ center3D,
              float* __restrict__ pts) {
    int bj = blockIdx.x;          // 0 .. B*J-1
    int b  = bj / J_;
    const float* vol = hf + (long long)bj * D3_;

    float s = 0.f, sx = 0.f, sy = 0.f, sz = 0.f;
    for (int t = threadIdx.x; t < D3_; t += 256) {
        float v  = vol[t];
        int d2   = t % D_;
        int rr   = t / D_;
        int d1   = rr % D_;
        int d0   = rr / D_;
        s  += v;
        sx += v * (float)d0;
        sy += v * (float)d1;
        sz += v * (float)d2;
    }

    __shared__ float part[4][256];
    part[0][threadIdx.x] = s;
    part[1][threadIdx.x] = sx;
    part[2][threadIdx.x] = sy;
    part[3][threadIdx.x] = sz;
    __syncthreads();

    if (threadIdx.x < 32) {       // whole wave 0 enters -> EXEC all ones
        int lane = threadIdx.x;
        v2f ones;
        ones.x = 1.0f;
        ones.y = 1.0f;

        float totals[4];
#pragma unroll
        for (int q = 0; q < 4; ++q) {
            v8f acc = {};
#pragma unroll
            for (int c = 0; c < 4; ++c) {
                v2f bm;
                bm.x = part[q][c * 64 + lane];
                bm.y = part[q][c * 64 + 32 + lane];
                // D = ones(16x4) * bm(4x16) + acc  -> column sums accumulate
                acc = __builtin_amdgcn_wmma_f32_16x16x4_f32(
                    /*neg_a=*/false, ones, /*neg_b=*/false, bm,
                    /*c_mod=*/(short)0, acc, /*reuse_a=*/false, /*reuse_b=*/false);
            }
            // D row M=0 lives in element 0 of lanes 0..15; sum those lanes.
            float r = acc[0];
            r += __shfl_down(r, 8, 32);
            r += __shfl_down(r, 4, 32);
            r += __shfl_down(r, 2, 32);
            r += __shfl_down(r, 1, 32);
            totals[q] = r;        // valid on lane 0
        }

        if (lane == 0) {
            float inv = 1.0f / totals[0];
            // points3D = p * GRID_SPACING * 2 - GRID_SIZE/GRID_SPACING + center3D
            //          = p * 4 - 100 + center3D
            pts[bj * 3 + 0] = totals[1] * inv * 4.0f - 100.0f + center3D[b * 3 + 0];
            pts[bj * 3 + 1] = totals[2] * inv * 4.0f - 100.0f + center3D[b * 3 + 1];
            pts[bj * 3 + 2] = totals[3] * inv * 4.0f - 100.0f + center3D[b * 3 + 2];
        }
    }
}

// ---------------------------------------------------------------------
extern "C" void kernel_launch(void* const* d_in, const int* in_sizes, int n_in,
                              void* d_out, int out_size, void* d_ws, size_t ws_size,
                              hipStream_t stream) {
    const float* heatmaps_batch = (const float*)d_in[0]; // [B,NCAM,J,HM,HM]
    const float* heatmap_vol    = (const float*)d_in[1]; // [B,J,D,D,D]
    const float* center3D       = (const float*)d_in[2]; // [B,1,3]
    const int*   centerHM       = (const int*)d_in[3];   // [B,NCAM,2]

    float* out = (float*)d_out;
    float* hf  = out;                 // heatmap_final   (5,750,000 f32)
    float* pad = out + HF_SIZE;       // heatmaps_padded (90,439,680 f32), 16B aligned
    float* pts = out + PTS_OFF;       // points3D        (138 f32)

    // Kernel 1: canvas placement (dominant: ~362 MB stores)
    long long total4 = PAD_SIZE / 4;
    int blocks1 = (int)((total4 + 255) / 256);
    place_kernel<<<blocks1, 256, 0, stream>>>(heatmaps_batch, centerHM, pad);

    // Kernel 2: softplus (HF_SIZE divisible by 4)
    int n4 = (int)(HF_SIZE / 4);
    softplus_kernel<<<(n4 + 255) / 256, 256, 0, stream>>>(
        (const float4*)heatmap_vol, (float4*)hf, n4);

    // Kernel 3: fused soft-argmax (reads hf written above; same stream order)
    reduce_kernel<<<B_ * J_, 256, 0, stream>>>(hf, center3D, pts);
}